// DIEN_84791244357900
// MI455X (gfx1250) — compile-verified
//
#include <hip/hip_runtime.h>
#include <hip/hip_bf16.h>
#include <math.h>

// ---------------------------------------------------------------------------
// DIEN forward for MI455X (gfx1250, wave32, WMMA).
// All fp32, using V_WMMA_F32_16X16X4_F32 (A:16x4, B:4x16, C:16x16 per wave).
//
// VGPR layouts (CDNA5 ISA 7.12.2):
//   A (16x4 f32, 2 VGPR): lanes 0-15 rows M=0..15 hold K={k0,k0+1};
//                         lanes 16-31 same rows hold K={k0+2,k0+3}.
//   B (4x16 f32, 2 VGPR): lanes 0-15 cols N=0..15 hold K rows {k0,k0+1};
//                         lanes 16-31 hold K rows {k0+2,k0+3}.
//   C (16x16 f32, 8 VGPR): vgpr v, lane l -> row = v + 8*(l>=16), col = l&15.
// ---------------------------------------------------------------------------

typedef __attribute__((ext_vector_type(2))) float v2f;
typedef __attribute__((ext_vector_type(8))) float v8f;

#define BB 512
#define TT 200
#define DD 64
#define N3 192
#define WHS 200   // Wh LDS stride (2*200 % 64 == 16 -> halves hit disjoint banks)
#define HS  66    // h / rh LDS stride

__device__ __forceinline__ v8f wmma_f32(v2f a, v2f b, v8f c) {
  return __builtin_amdgcn_wmma_f32_16x16x4_f32(false, a, false, b, (short)0, c,
                                               false, false);
}
__device__ __forceinline__ float sigmoidf_(float x) {
  return 1.0f / (1.0f + expf(-x));
}

// ---------------------------------------------------------------------------
// Gather user / target embeddings.
// ---------------------------------------------------------------------------
__global__ __launch_bounds__(256) void gather_kernel(
    const float* __restrict__ emb_user, const float* __restrict__ emb_item,
    const int* __restrict__ user_ids, const int* __restrict__ tgt_ids,
    float* __restrict__ user_e, float* __restrict__ tgt_e) {
  int i = blockIdx.x * blockDim.x + threadIdx.x;
  if (i < BB * DD) {
    int r = i >> 6, c = i & 63;
    user_e[i] = emb_user[(size_t)user_ids[r] * DD + c];
    tgt_e[i]  = emb_item[(size_t)tgt_ids[r] * DD + c];
  }
}

// ---------------------------------------------------------------------------
// xg = A @ W[64,192] + bias.  A row r is either emb_item[ids[r]] (gathered)
// or Asrc[r] (dense).  Block: 128 threads = 4 waves; tile 16 rows x 64 cols,
// wave w owns 16-col subtile.  Grid: (M/16, 192/64).
// ---------------------------------------------------------------------------
__global__ __launch_bounds__(128) void gemm_xg_kernel(
    const float* __restrict__ Asrc, const int* __restrict__ ids,
    const float* __restrict__ emb, const float* __restrict__ W,
    const float* __restrict__ bias, float* __restrict__ C) {
  __shared__ alignas(16) float As[16 * 68];
  __shared__ alignas(16) float Ws[64 * 68];
  int tid = threadIdx.x, wave = tid >> 5, lane = tid & 31;
  int half = lane >> 4, l16 = lane & 15;
  int rbase = blockIdx.x * 16;
  int cbase = blockIdx.y * 64;

  for (int i = tid; i < 16 * 64; i += 128) {
    int r = i >> 6, c = i & 63;
    int row = rbase + r;
    const float* src =
        ids ? (emb + (size_t)ids[row] * DD) : (Asrc + (size_t)row * DD);
    As[r * 68 + c] = src[c];
  }
  for (int i = tid; i < 64 * 64; i += 128) {
    int r = i >> 6, c = i & 63;
    Ws[r * 68 + c] = W[r * N3 + cbase + c];
  }
  __syncthreads();

  v8f acc = {};
  int ct = wave * 16;  // column subtile within the 64-col panel
#pragma unroll
  for (int k0 = 0; k0 < 64; k0 += 4) {
    v2f a = *(const v2f*)&As[l16 * 68 + k0 + 2 * half];
    v2f b;
    b.x = Ws[(k0 + 2 * half) * 68 + ct + l16];
    b.y = Ws[(k0 + 2 * half + 1) * 68 + ct + l16];
    acc = wmma_f32(a, b, acc);
  }
  float bb = bias[cbase + ct + l16];
#pragma unroll
  for (int v = 0; v < 8; ++v) {
    int row = rbase + v + 8 * half;
    C[(size_t)row * N3 + cbase + ct + l16] = acc[v] + bb;
  }
}

// ---------------------------------------------------------------------------
// Recurrent scan (GRU mode=0 writes all hidden states; AUGRU mode=1 applies
// attention to the update gate and writes only the final hidden state).
// Block: 128 threads = 4 waves covering 16 batch rows; wave w computes gate
// column tile 16w of z, r, n.  h and r*h live in LDS shared by the 4 waves.
// Grid: 32 blocks (512 rows / 16).
// ---------------------------------------------------------------------------
__global__ __launch_bounds__(128) void rnn_scan_kernel(
    const float* __restrict__ xg,   // [B*T, 192] precomputed x@Wx + b
    const float* __restrict__ Wh,   // [64, 192]
    const float* __restrict__ att,  // [B, T] (mode 1 only)
    float* __restrict__ out,        // mode0: [B*T,64]; mode1: [B,64]
    int mode) {
  __shared__ alignas(16) float Wh_s[64 * WHS];
  __shared__ alignas(16) float h_s[16 * HS];
  __shared__ alignas(16) float rh_s[16 * HS];
  int tid = threadIdx.x, wave = tid >> 5, lane = tid & 31;
  int half = lane >> 4, l16 = lane & 15;

  for (int i = tid; i < 64 * N3; i += 128) {
    int r = i / N3, c = i - r * N3;
    Wh_s[r * WHS + c] = Wh[i];
  }
  for (int i = tid; i < 16 * HS; i += 128) { h_s[i] = 0.f; rh_s[i] = 0.f; }
  __syncthreads();

  int rowbase = blockIdx.x * 16;
  int cz = 16 * wave;        // z gate columns   (Wh[:, 0:64])
  int cr = 64 + 16 * wave;   // r gate columns   (Wh[:, 64:128])
  int cn = 128 + 16 * wave;  // n gate columns   (Wh[:, 128:192])

  for (int t = 0; t < TT; ++t) {
    // --- hg_z, hg_r = h @ Wh[:, :128] (this wave's 16-col slices) ---
    v8f accz = {}, accr = {};
#pragma unroll
    for (int k0 = 0; k0 < DD; k0 += 4) {
      v2f a = *(const v2f*)&h_s[l16 * HS + k0 + 2 * half];
      v2f bz, br;
      bz.x = Wh_s[(k0 + 2 * half) * WHS + cz + l16];
      bz.y = Wh_s[(k0 + 2 * half + 1) * WHS + cz + l16];
      br.x = Wh_s[(k0 + 2 * half) * WHS + cr + l16];
      br.y = Wh_s[(k0 + 2 * half + 1) * WHS + cr + l16];
      accz = wmma_f32(a, bz, accz);
      accr = wmma_f32(a, br, accr);
    }
    // --- gates, write r*h into LDS ---
    float whc[8];  // coefficient on old h in the blend
#pragma unroll
    for (int v = 0; v < 8; ++v) {
      int row = v + 8 * half;
      int rb = rowbase + row;
      size_t xoff = ((size_t)rb * TT + t) * N3;
      if (t + 1 < TT) __builtin_prefetch(&xg[xoff + N3 + cz + l16], 0, 1);
      float z = sigmoidf_(xg[xoff + cz + l16] + accz[v]);
      float r = sigmoidf_(xg[xoff + cr + l16] + accr[v]);
      if (mode) {
        float u = att[(size_t)rb * TT + t] * z;
        whc[v] = 1.0f - u;          // AUGRU: h' = (1-u)h + u n
      } else {
        whc[v] = z;                 // GRU:   h' = z h + (1-z) n
      }
      rh_s[row * HS + cz + l16] = r * h_s[row * HS + cz + l16];
    }
    __syncthreads();  // all waves' r*h visible before n matmul

    // --- hg_n = (r*h) @ Wh[:, 128:] ---
    v8f accn = {};
#pragma unroll
    for (int k0 = 0; k0 < DD; k0 += 4) {
      v2f a = *(const v2f*)&rh_s[l16 * HS + k0 + 2 * half];
      v2f b;
      b.x = Wh_s[(k0 + 2 * half) * WHS + cn + l16];
      b.y = Wh_s[(k0 + 2 * half + 1) * WHS + cn + l16];
      accn = wmma_f32(a, b, accn);
    }
    // --- blend and update h (each wave owns disjoint h columns) ---
#pragma unroll
    for (int v = 0; v < 8; ++v) {
      int row = v + 8 * half;
      int rb = rowbase + row;
      size_t xoff = ((size_t)rb * TT + t) * N3;
      float n = tanhf(xg[xoff + cn + l16] + accn[v]);
      float hold = h_s[row * HS + cz + l16];
      float hn = whc[v] * hold + (1.0f - whc[v]) * n;
      h_s[row * HS + cz + l16] = hn;
      if (mode == 0)
        out[((size_t)rb * TT + t) * DD + cz + l16] = hn;
      else if (t == TT - 1)
        out[(size_t)rb * DD + cz + l16] = hn;
    }
    __syncthreads();  // h update visible to all waves before next step
  }
}

// ---------------------------------------------------------------------------
// Attention: scores[b,t] = <tgt[b], gru_out[b,t]>; softmax over t.
// One block per batch row.
// ---------------------------------------------------------------------------
__global__ __launch_bounds__(256) void attn_kernel(
    const float* __restrict__ gru_out, const float* __restrict__ tgt,
    float* __restrict__ att) {
  __shared__ float tg[DD];
  __shared__ float sc[256];
  __shared__ float red[256];
  int b = blockIdx.x, tid = threadIdx.x;
  if (tid < DD) tg[tid] = tgt[b * DD + tid];
  __syncthreads();
  float s = -3.0e38f;
  if (tid < TT) {
    const float* g = gru_out + ((size_t)b * TT + tid) * DD;
    float acc = 0.f;
#pragma unroll
    for (int d = 0; d < DD; ++d) acc += tg[d] * g[d];
    sc[tid] = acc;
    s = acc;
  }
  red[tid] = s;
  __syncthreads();
  for (int off = 128; off >= 1; off >>= 1) {
    if (tid < off) red[tid] = fmaxf(red[tid], red[tid + off]);
    __syncthreads();
  }
  float mx = red[0];
  __syncthreads();
  float e = (tid < TT) ? expf(sc[tid] - mx) : 0.f;
  red[tid] = e;
  __syncthreads();
  for (int off = 128; off >= 1; off >>= 1) {
    if (tid < off) red[tid] += red[tid + off];
    __syncthreads();
  }
  float inv = 1.0f / red[0];
  if (tid < TT) att[(size_t)b * TT + tid] = e * inv;
}

// ---------------------------------------------------------------------------
// Auxiliary MLP + loss, fully fused.  Per 16-row tile:
//   A1[16,128]=concat(gru_out,beh_emb) -> h1=relu(A1@W1+b1)[16,112pad]
//   -> h2=relu(h1@W2+b2)[16,64pad] -> logits[16,2] -> -mean(log_softmax[...,0])
// Block: 64 threads = 2 waves = 32 rows.  Grid: 3200.  W1/W2 staged in LDS.
// ---------------------------------------------------------------------------
#define AUXS1 116  // >= 112 padded N of layer1, also h1 stride
#define AUXS2 68
__global__ __launch_bounds__(64) void aux_kernel(
    const float* __restrict__ gru_out, const int* __restrict__ beh_ids,
    const float* __restrict__ emb, const float* __restrict__ W1,
    const float* __restrict__ b1, const float* __restrict__ W2,
    const float* __restrict__ b2, const float* __restrict__ W3,
    const float* __restrict__ b3, float* __restrict__ loss_acc) {
  extern __shared__ float smem[];
  float* W1s = smem;                      // 128 * AUXS1
  float* W2s = W1s + 128 * AUXS1;         // 100 * AUXS2
  float* A1s = W2s + 100 * AUXS2;         // 2 waves * 16*132
  float* H1s = A1s + 2 * 16 * 132;        // 2 waves * 16*AUXS1

  int tid = threadIdx.x, wave = tid >> 5, lane = tid & 31;
  int half = lane >> 4, l16 = lane & 15;

  for (int i = tid; i < 128 * 112; i += 64) {
    int r = i / 112, c = i - r * 112;
    W1s[r * AUXS1 + c] = (c < 100) ? W1[r * 100 + c] : 0.f;
  }
  for (int i = tid; i < 100 * 64; i += 64) {
    int r = i >> 6, c = i & 63;
    W2s[r * AUXS2 + c] = (c < 50) ? W2[r * 50 + c] : 0.f;
  }
  float* A1 = A1s + wave * 16 * 132;
  float* H1 = H1s + wave * 16 * AUXS1;
  int rowbase = blockIdx.x * 32 + wave * 16;

  for (int i = lane; i < 16 * 128; i += 32) {
    int r = i >> 7, c = i & 127;
    int row = rowbase + r;
    A1[r * 132 + c] = (c < 64)
                          ? gru_out[(size_t)row * DD + c]
                          : emb[(size_t)beh_ids[row] * DD + (c - 64)];
  }
  __syncthreads();

  // layer 1: K=128, 7 column tiles (N padded to 112)
  for (int j = 0; j < 7; ++j) {
    v8f acc = {};
#pragma unroll
    for (int k0 = 0; k0 < 128; k0 += 4) {
      v2f a = *(const v2f*)&A1[l16 * 132 + k0 + 2 * half];
      v2f b;
      b.x = W1s[(k0 + 2 * half) * AUXS1 + 16 * j + l16];
      b.y = W1s[(k0 + 2 * half + 1) * AUXS1 + 16 * j + l16];
      acc = wmma_f32(a, b, acc);
    }
    int cb = 16 * j + l16;
    float bb = (cb < 100) ? b1[cb] : 0.f;
#pragma unroll
    for (int v = 0; v < 8; ++v) {
      float h = acc[v] + bb;
      H1[(v + 8 * half) * AUXS1 + cb] = h > 0.f ? h : 0.f;
    }
  }
  __syncthreads();

  // layer 2: K=100, 4 column tiles (N padded to 64)
  v8f h2f[4];
  for (int j = 0; j < 4; ++j) {
    v8f acc = {};
#pragma unroll
    for (int k0 = 0; k0 < 100; k0 += 4) {
      v2f a = *(const v2f*)&H1[l16 * AUXS1 + k0 + 2 * half];
      v2f b;
      b.x = W2s[(k0 + 2 * half) * AUXS2 + 16 * j + l16];
      b.y = W2s[(k0 + 2 * half + 1) * AUXS2 + 16 * j + l16];
      acc = wmma_f32(a, b, acc);
    }
    int cb = 16 * j + l16;
    float bb = (cb < 50) ? b2[cb] : 0.f;
#pragma unroll
    for (int v = 0; v < 8; ++v) {
      float h = acc[v] + bb;
      h2f[j][v] = h > 0.f ? h : 0.f;
    }
  }
  // stash h2 into (now dead) A1 region so each lane can read a full row
  for (int j = 0; j < 4; ++j)
#pragma unroll
    for (int v = 0; v < 8; ++v)
      A1[(v + 8 * half) * 132 + 16 * j + l16] = h2f[j][v];
  __syncthreads();

  // logits: lane = row (0..15) + 16*class
  int r = lane & 15, cls = lane >> 4;
  float lg = b3[cls];
  for (int k = 0; k < 50; ++k) lg += A1[r * 132 + k] * W3[k * 2 + cls];
  float other = __shfl(lg, lane ^ 16, 32);
  float l0 = half ? other : lg;
  float l1 = half ? lg : other;
  float m = fmaxf(l0, l1);
  float lse = m + logf(expf(l0 - m) + expf(l1 - m));
  float term = (half == 0) ? (lse - l0) : 0.f;  // -log_softmax[...,0]
  for (int off = 8; off >= 1; off >>= 1) term += __shfl_down(term, off, 32);
  if (lane == 0) atomicAdd(loss_acc, term * (1.0f / (512.0f * 200.0f)));
}

// ---------------------------------------------------------------------------
// BatchNorm statistics over batch for join = [h_final, user_emb] (128 feats).
// ---------------------------------------------------------------------------
__global__ __launch_bounds__(128) void bn_stats_kernel(
    const float* __restrict__ hf, const float* __restrict__ ue,
    float* __restrict__ mu, float* __restrict__ var) {
  int f = threadIdx.x;
  const float* src = (f < DD) ? (hf + f) : (ue + (f - DD));
  float s = 0.f, s2 = 0.f;
  for (int r = 0; r < BB; ++r) {
    float v = src[(size_t)r * DD];
    s += v;
    s2 += v * v;
  }
  float m = s * (1.0f / (float)BB);
  mu[f] = m;
  var[f] = s2 * (1.0f / (float)BB) - m * m;
}

// ---------------------------------------------------------------------------
// Final head: BN -> relu(fc1) -> relu(fc2) -> fc3 -> softmax.
// (prelu(relu(x)) == relu(x) since prelu only acts on negatives.)
// One block per batch row.
// ---------------------------------------------------------------------------
__global__ __launch_bounds__(64) void head_kernel(
    const float* __restrict__ hf, const float* __restrict__ ue,
    const float* __restrict__ mu, const float* __restrict__ var,
    const float* __restrict__ gamma, const float* __restrict__ beta,
    const float* __restrict__ W1, const float* __restrict__ b1,
    const float* __restrict__ W2, const float* __restrict__ b2,
    const float* __restrict__ W3, const float* __restrict__ b3,
    float* __restrict__ out) {
  __shared__ float xb[128];
  __shared__ float h1[200];
  __shared__ float h2[80];
  __shared__ float lg[2];
  int r = blockIdx.x, tid = threadIdx.x;
  for (int f = tid; f < 128; f += 64) {
    float j = (f < DD) ? hf[(size_t)r * DD + f] : ue[(size_t)r * DD + f - DD];
    xb[f] = (j - mu[f]) * rsqrtf(var[f] + 1e-3f) * gamma[f] + beta[f];
  }
  __syncthreads();
  for (int j = tid; j < 200; j += 64) {
    float a = b1[j];
    for (int k = 0; k < 128; ++k) a += xb[k] * W1[k * 200 + j];
    h1[j] = a > 0.f ? a : 0.f;
  }
  __syncthreads();
  for (int j = tid; j < 80; j += 64) {
    float a = b2[j];
    for (int k = 0; k < 200; ++k) a += h1[k] * W2[k * 80 + j];
    h2[j] = a > 0.f ? a : 0.f;
  }
  __syncthreads();
  if (tid < 2) {
    float a = b3[tid];
    for (int k = 0; k < 80; ++k) a += h2[k] * W3[k * 2 + tid];
    lg[tid] = a;
    out[1024 + r * 2 + tid] = a;  // logit
  }
  __syncthreads();
  if (tid == 0) {
    float l0 = lg[0], l1 = lg[1];
    float m = fmaxf(l0, l1);
    float e0 = expf(l0 - m), e1 = expf(l1 - m);
    float inv = 1.0f / (e0 + e1);
    out[r * 2 + 0] = e0 * inv;  // softmax output
    out[r * 2 + 1] = e1 * inv;
  }
}

// ---------------------------------------------------------------------------
extern "C" void kernel_launch(void* const* d_in, const int* in_sizes, int n_in,
                              void* d_out, int out_size, void* d_ws,
                              size_t ws_size, hipStream_t stream) {
  (void)in_sizes; (void)n_in; (void)out_size; (void)ws_size;
  const int* user_ids  = (const int*)d_in[0];
  const int* beh_ids   = (const int*)d_in[1];
  const int* tgt_ids   = (const int*)d_in[2];
  const float* emb_user = (const float*)d_in[3];
  const float* emb_item = (const float*)d_in[4];
  const float* gru_Wx = (const float*)d_in[5];
  const float* gru_Wh = (const float*)d_in[6];
  const float* gru_b  = (const float*)d_in[7];
  const float* aug_Wx = (const float*)d_in[8];
  const float* aug_Wh = (const float*)d_in[9];
  const float* aug_b  = (const float*)d_in[10];
  const float* aux_W1 = (const float*)d_in[11];
  const float* aux_b1 = (const float*)d_in[12];
  const float* aux_W2 = (const float*)d_in[13];
  const float* aux_b2 = (const float*)d_in[14];
  const float* aux_W3 = (const float*)d_in[15];
  const float* aux_b3 = (const float*)d_in[16];
  const float* bn_gamma = (const float*)d_in[17];
  const float* bn_beta  = (const float*)d_in[18];
  const float* fc_W1 = (const float*)d_in[19];
  const float* fc_b1 = (const float*)d_in[20];
  // d_in[21] = prelu_a1: prelu(relu(x)) == relu(x), unused
  const float* fc_W2 = (const float*)d_in[22];
  const float* fc_b2 = (const float*)d_in[23];
  // d_in[24] = prelu_a2: unused
  const float* fc_W3 = (const float*)d_in[25];
  const float* fc_b3 = (const float*)d_in[26];

  // workspace layout (floats), ~106 MB total
  float* ws = (float*)d_ws;
  float* xg      = ws;                                   // [B*T,192] (reused)
  float* gru_out = xg + (size_t)BB * TT * N3;            // [B*T,64]
  float* att     = gru_out + (size_t)BB * TT * DD;       // [B,T]
  float* user_e  = att + (size_t)BB * TT;                // [B,64]
  float* tgt_e   = user_e + BB * DD;                     // [B,64]
  float* hfin    = tgt_e + BB * DD;                      // [B,64]
  float* mu      = hfin + BB * DD;                       // [128]
  float* var     = mu + 128;                             // [128]

  float* out = (float*)d_out;  // [0,1024)=softmax, [1024,2048)=logit, [2048]=aux
  hipMemsetAsync(out + 2048, 0, sizeof(float), stream);

  gather_kernel<<<(BB * DD + 255) / 256, 256, 0, stream>>>(
      emb_user, emb_item, user_ids, tgt_ids, user_e, tgt_e);

  // GRU input projections: gather(beh) @ gru_Wx + b
  gemm_xg_kernel<<<dim3((BB * TT) / 16, N3 / 64), 128, 0, stream>>>(
      nullptr, beh_ids, emb_item, gru_Wx, gru_b, xg);
  rnn_scan_kernel<<<BB / 16, 128, 0, stream>>>(xg, gru_Wh, nullptr, gru_out, 0);

  attn_kernel<<<BB, 256, 0, stream>>>(gru_out, tgt_e, att);

  // AUGRU input projections: gru_out @ aug_Wx + b (reuse xg buffer)
  gemm_xg_kernel<<<dim3((BB * TT) / 16, N3 / 64), 128, 0, stream>>>(
      gru_out, nullptr, nullptr, aug_Wx, aug_b, xg);
  rnn_scan_kernel<<<BB / 16, 128, 0, stream>>>(xg, aug_Wh, att, hfin, 1);

  size_t aux_lds =
      (size_t)(128 * AUXS1 + 100 * AUXS2 + 2 * 16 * 132 + 2 * 16 * AUXS1) *
      sizeof(float);
  (void)hipFuncSetAttribute(reinterpret_cast<const void*>(aux_kernel),
                            hipFuncAttributeMaxDynamicSharedMemorySize,
                            (int)aux_lds);
  aux_kernel<<<(BB * TT) / 32, 64, aux_lds, stream>>>(
      gru_out, beh_ids, emb_item, aux_W1, aux_b1, aux_W2, aux_b2, aux_W3,
      aux_b3, out + 2048);

  bn_stats_kernel<<<1, 128, 0, stream>>>(hfin, user_e, mu, var);
  head_kernel<<<BB, 64, 0, stream>>>(hfin, user_e, mu, var, bn_gamma, bn_beta,
                                     fc_W1, fc_b1, fc_W2, fc_b2, fc_W3, fc_b3,
                                     out);
}